// DVGO_53841710023282
// MI455X (gfx1250) — compile-verified
//
#include <hip/hip_runtime.h>
#include <hip/hip_bf16.h>
#include <math.h>

typedef float v2f __attribute__((ext_vector_type(2)));
typedef float v8f __attribute__((ext_vector_type(8)));

namespace {
constexpr int   R    = 8192;
constexpr int   S    = 558;
constexpr int   SP1  = S + 1;
constexpr int   S_FULL = (S / 16) * 16;          // 544: last full-chunk boundary
constexpr int   G    = 160;
constexpr size_t N3  = (size_t)G * G * G;
constexpr float NEARF     = 0.05f;
constexpr float FARF      = 6.0f;
constexpr float STEP_DIST = 0.00625f;            // STEPSIZE * VOXEL_SIZE = 0.5 * (2/160)
constexpr float ACT_SHIFT = -13.8155095579647f;  // log(1/(1-1e-6) - 1)

// Flat output layout (reference tuple order, float32):
constexpr size_t O_AIC  = 0;                          // alphainv_cum [R, S+1]
constexpr size_t O_W    = (size_t)R * SP1;            // weights      [R, S]   (alpha staged here by kernel 1)
constexpr size_t O_LAST = O_W + (size_t)R * S;        // last alphainv [R, 1]
constexpr size_t O_RGB  = O_LAST + (size_t)R;         // rgb          [R, S, 3]
constexpr size_t O_RGBM = O_RGB + (size_t)R * S * 3;  // rgb_marched  [R, 3]
} // namespace

__device__ __forceinline__ float sigmoidf_(float x) { return 1.0f / (1.0f + expf(-x)); }

// ---------------------------------------------------------------------------
// Kernel 1: per (ray, sample) point generation + trilinear gathers.
// Grids (~115 MB total) are L2-resident on MI455X (192 MB L2), so the 8-corner
// gathers hit L2. Ray-uniform data (blockIdx.y = ray) lowers to scalar loads.
// Writes rgb to its final region; stages alpha in the weights region.
// ---------------------------------------------------------------------------
__global__ void __launch_bounds__(192) dvgo_sample_kernel(
    const float* __restrict__ rays_o, const float* __restrict__ rays_d,
    const float* __restrict__ jitter, const int* __restrict__ em_modes,
    const float* __restrict__ density, const float* __restrict__ off_color,
    const float* __restrict__ emo_color, float* __restrict__ out)
{
    const int r = blockIdx.y;
    const int s = blockIdx.x * blockDim.x + threadIdx.x;
    if (s >= S) return;

    const float ox = rays_o[3*r+0], oy = rays_o[3*r+1], oz = rays_o[3*r+2];
    const float dx = rays_d[3*r+0], dy = rays_d[3*r+1], dz = rays_d[3*r+2];
    const float jit = jitter[r];
    const bool  on  = (em_modes[r] == 1);

    const float vx = (dx == 0.f) ? 1e-6f : dx;
    const float vy = (dy == 0.f) ? 1e-6f : dy;
    const float vz = (dz == 0.f) ? 1e-6f : dz;
    const float rax = ( 1.f - ox) / vx, ray2 = ( 1.f - oy) / vy, raz = ( 1.f - oz) / vz;
    const float rbx = (-1.f - ox) / vx, rby  = (-1.f - oy) / vy, rbz = (-1.f - oz) / vz;
    float tmin = fmaxf(fmaxf(fminf(rax, rbx), fminf(ray2, rby)), fminf(raz, rbz));
    float tmax = fminf(fminf(fmaxf(rax, rbx), fmaxf(ray2, rby)), fmaxf(raz, rbz));
    tmin = fminf(fmaxf(tmin, NEARF), FARF);
    tmax = fminf(fmaxf(tmax, NEARF), FARF);
    bool mask = (tmax <= tmin);

    const float dn      = sqrtf(dx*dx + dy*dy + dz*dz);
    const float interpx = tmin + STEP_DIST * ((float)s + jit) / dn;
    const float px = ox + dx * interpx;
    const float py = oy + dy * interpx;
    const float pz = oz + dz * interpx;
    mask = mask || (px < -1.f) || (px > 1.f) || (py < -1.f) || (py > 1.f)
                || (pz < -1.f) || (pz > 1.f);

    // Voxel-space coords, align_corners=True: t01 * (G-1)
    const float tx = (px + 1.f) * 0.5f * 159.f;
    const float ty = (py + 1.f) * 0.5f * 159.f;
    const float tz = (pz + 1.f) * 0.5f * 159.f;
    const float fx0 = floorf(tx), fy0 = floorf(ty), fz0 = floorf(tz);
    const int ix = (int)fx0, iy = (int)fy0, iz = (int)fz0;
    const float fx = tx - fx0, fy = ty - fy0, fz = tz - fz0;
    const float wx[2] = {1.f - fx, fx};
    const float wy[2] = {1.f - fy, fy};
    const float wz[2] = {1.f - fz, fz};
    const int cx[2] = {ix, ix + 1};
    const int cy[2] = {iy, iy + 1};
    const int cz[2] = {iz, iz + 1};

    float dens = 0.f;
    float oc0 = 0.f, oc1 = 0.f, oc2 = 0.f;
    float ec0 = 0.f, ec1 = 0.f, ec2 = 0.f;
    #pragma unroll
    for (int a = 0; a < 2; ++a)
    #pragma unroll
    for (int b = 0; b < 2; ++b)
    #pragma unroll
    for (int c = 0; c < 2; ++c) {
        const int X = cx[a], Y = cy[b], Z = cz[c];
        if ((unsigned)X < (unsigned)G && (unsigned)Y < (unsigned)G && (unsigned)Z < (unsigned)G) {
            const float w = wx[a] * wy[b] * wz[c];
            const size_t idx = ((size_t)X * G + (size_t)Y) * G + (size_t)Z;
            dens += w * density[idx];
            oc0  += w * off_color[idx];
            oc1  += w * off_color[N3 + idx];
            oc2  += w * off_color[2*N3 + idx];
            ec0  += w * emo_color[idx];
            ec1  += w * emo_color[N3 + idx];
            ec2  += w * emo_color[2*N3 + idx];
        }
    }

    // alpha = 1 - exp(-softplus(dens + shift) * 0.5), zeroed where masked
    const float xs = dens + ACT_SHIFT;
    const float sp = (xs > 0.f) ? (xs + log1pf(expf(-xs))) : log1pf(expf(xs));
    const float alpha = mask ? 0.f : (1.f - expf(-sp * 0.5f));
    out[O_W + (size_t)r * S + s] = alpha;   // staged; kernel 2 overwrites with weights

    float* rgbp = out + O_RGB + ((size_t)r * S + s) * 3u;
    rgbp[0] = (on ? sigmoidf_(ec0) : 0.f) + sigmoidf_(oc0);
    rgbp[1] = (on ? sigmoidf_(ec1) : 0.f) + sigmoidf_(oc1);
    rgbp[2] = (on ? sigmoidf_(ec2) : 0.f) + sigmoidf_(oc2);
}

// ---------------------------------------------------------------------------
// Kernel 2: per-ray scan. One wave32 owns 16 rays. For each 16-sample chunk,
// the inclusive log-space prefix sum of 16 rays x 16 samples is computed as
// Y = X * U (U upper-triangular ones) with FOUR V_WMMA_F32_16X16X4_F32
// accumulations. Carries propagate across chunks via a lane-15/31 broadcast.
//   A (16x4 f32): lane%16 = ray (M); half-wave selects K {0,1} vs {2,3}.
//   C/D:          VGPR j = ray j (+8 for upper half); lane%16 = sample (N).
// Fast path: 34 full chunks with unguarded 8-byte-aligned paired loads
// (element offset rayA*S + s0 + 4kc + 2h is provably even). Guarded tail
// handles the final partial chunk (s0 = 544, 14 valid samples).
// NOTE: deliberately no __restrict__ here — alpha reads alias weight writes.
// ---------------------------------------------------------------------------
__global__ void __launch_bounds__(256) dvgo_scan_kernel(float* out)
{
    const int lane = threadIdx.x & 31;
    const int wid  = blockIdx.x * (blockDim.x >> 5) + (threadIdx.x >> 5);
    const int r0   = wid * 16;
    if (r0 >= R) return;                 // wave-uniform exit: EXEC stays all-1s for WMMA
    const int h = lane >> 4;             // half-wave
    const int n = lane & 15;             // sample position within chunk / ray-in-half

    const float* alphaBuf = out + O_W;   // aliases the weights region (intentional)

    float carry[8];
    float acc0[8], acc1[8], acc2[8];
    #pragma unroll
    for (int j = 0; j < 8; ++j) { carry[j] = 0.f; acc0[j] = acc1[j] = acc2[j] = 0.f; }

    if (n == 0) {
        #pragma unroll
        for (int j = 0; j < 8; ++j)
            out[O_AIC + (size_t)(r0 + j + 8*h) * SP1] = 1.0f;  // alphainv_cum[:,0] = 1
    }

    // Loop-invariant B fragments: U[k, n] = (global k <= n)
    v2f Bf[4];
    #pragma unroll
    for (int kc = 0; kc < 4; ++kc) {
        Bf[kc].x = ((4*kc + 2*h + 0) <= n) ? 1.f : 0.f;
        Bf[kc].y = ((4*kc + 2*h + 1) <= n) ? 1.f : 0.f;
    }

    const int rayA = r0 + n;             // ray this lane feeds into the A fragment
    const int src  = lane | 15;          // carry broadcast source (15 or 31)

    // ---- fast path: full chunks, no bounds checks ----
    for (int s0 = 0; s0 < S_FULL; s0 += 16) {
        v8f Y = {0.f, 0.f, 0.f, 0.f, 0.f, 0.f, 0.f, 0.f};
        #pragma unroll
        for (int kc = 0; kc < 4; ++kc) {
            const float2 ap =
                *(const float2*)(alphaBuf + (size_t)rayA * S + (size_t)(s0 + 4*kc + 2*h));
            v2f A;
            A.x = logf(fmaxf(1.f - ap.x, 1e-10f));
            A.y = logf(fmaxf(1.f - ap.y, 1e-10f));
            Y = __builtin_amdgcn_wmma_f32_16x16x4_f32(false, A, false, Bf[kc],
                                                      (short)0, Y, false, false);
        }

        const int s = s0 + n;
        #pragma unroll
        for (int j = 0; j < 8; ++j) {
            const int   ray   = r0 + j + 8*h;
            const float alpha = alphaBuf[(size_t)ray * S + s];
            const float p     = fmaxf(1.f - alpha, 1e-10f);
            const float aicN  = expf(carry[j] + Y[j]);      // alphainv_cum[s+1]
            const float aicC  = aicN / p;                   // alphainv_cum[s]
            out[O_AIC + (size_t)ray * SP1 + (size_t)(s + 1)] = aicN;
            const float w = alpha * aicC;
            out[O_W + (size_t)ray * S + s] = w;             // overwrite staged alpha
            const float* rgbp = out + O_RGB + ((size_t)ray * S + s) * 3u;
            acc0[j] += w * rgbp[0];
            acc1[j] += w * rgbp[1];
            acc2[j] += w * rgbp[2];
        }

        #pragma unroll
        for (int j = 0; j < 8; ++j)
            carry[j] += __shfl(Y[j], src, 32);
    }

    // ---- tail chunk: s0 = S_FULL, samples 544..557 valid ----
    {
        const int s0 = S_FULL;
        v8f Y = {0.f, 0.f, 0.f, 0.f, 0.f, 0.f, 0.f, 0.f};
        #pragma unroll
        for (int kc = 0; kc < 4; ++kc) {
            const int sA = s0 + 4*kc + 2*h;
            const float a0 = (sA     < S) ? alphaBuf[(size_t)rayA * S + sA    ] : 0.f;
            const float a1 = (sA + 1 < S) ? alphaBuf[(size_t)rayA * S + sA + 1] : 0.f;
            v2f A;
            A.x = logf(fmaxf(1.f - a0, 1e-10f));
            A.y = logf(fmaxf(1.f - a1, 1e-10f));
            Y = __builtin_amdgcn_wmma_f32_16x16x4_f32(false, A, false, Bf[kc],
                                                      (short)0, Y, false, false);
        }

        const int  s  = s0 + n;
        const bool sv = (s < S);
        #pragma unroll
        for (int j = 0; j < 8; ++j) {
            if (sv) {
                const int   ray   = r0 + j + 8*h;
                const float alpha = alphaBuf[(size_t)ray * S + s];
                const float p     = fmaxf(1.f - alpha, 1e-10f);
                const float aicN  = expf(carry[j] + Y[j]);
                const float aicC  = aicN / p;
                out[O_AIC + (size_t)ray * SP1 + (size_t)(s + 1)] = aicN;
                const float w = alpha * aicC;
                out[O_W + (size_t)ray * S + s] = w;
                const float* rgbp = out + O_RGB + ((size_t)ray * S + s) * 3u;
                acc0[j] += w * rgbp[0];
                acc1[j] += w * rgbp[1];
                acc2[j] += w * rgbp[2];
            }
        }

        #pragma unroll
        for (int j = 0; j < 8; ++j)
            carry[j] += __shfl(Y[j], src, 32);
    }

    // Reduce rgb_marched within each 16-lane half; write final outputs.
    #pragma unroll
    for (int j = 0; j < 8; ++j) {
        float a0 = acc0[j], a1 = acc1[j], a2 = acc2[j];
        #pragma unroll
        for (int m = 1; m < 16; m <<= 1) {
            a0 += __shfl_xor(a0, m, 32);
            a1 += __shfl_xor(a1, m, 32);
            a2 += __shfl_xor(a2, m, 32);
        }
        if (n == 0) {
            const int ray = r0 + j + 8*h;
            out[O_LAST + (size_t)ray] = expf(carry[j]);
            float* rm = out + O_RGBM + (size_t)ray * 3u;
            rm[0] = a0; rm[1] = a1; rm[2] = a2;
        }
    }
}

extern "C" void kernel_launch(void* const* d_in, const int* in_sizes, int n_in,
                              void* d_out, int out_size, void* d_ws, size_t ws_size,
                              hipStream_t stream) {
    (void)in_sizes; (void)n_in; (void)out_size; (void)d_ws; (void)ws_size;
    const float* rays_o = (const float*)d_in[0];
    const float* rays_d = (const float*)d_in[1];
    const float* jitter = (const float*)d_in[2];
    const int*   em     = (const int*)  d_in[3];
    const float* dens   = (const float*)d_in[4];
    const float* offc   = (const float*)d_in[5];
    const float* emoc   = (const float*)d_in[6];
    float* out = (float*)d_out;

    dim3 b1(192, 1, 1), g1((S + 191) / 192, R, 1);
    dvgo_sample_kernel<<<g1, b1, 0, stream>>>(rays_o, rays_d, jitter, em,
                                              dens, offc, emoc, out);

    dim3 b2(256, 1, 1), g2(R / 16 / 8, 1, 1);   // 512 waves, 16 rays each
    dvgo_scan_kernel<<<g2, b2, 0, stream>>>(out);
}